// Attention_45148696216287
// MI455X (gfx1250) — compile-verified
//
#include <hip/hip_runtime.h>

#define HEADS     16
#define KV_HEADS  4
#define DHEAD     64
#define DIM       1024
#define BATCH     2
#define SEQ       2048
#define GQ        (HEADS / KV_HEADS)      // 4
#define QCOLS     (HEADS * DHEAD)         // 1024
#define KVCOLS    (2 * KV_HEADS * DHEAD)  // 512
#define TOTCOLS   (QCOLS + KVCOLS)        // 1536

typedef __attribute__((ext_vector_type(2))) float v2f;
typedef __attribute__((ext_vector_type(8))) float v8f;

static __device__ __forceinline__ v8f wmma_f32(v2f a, v2f b, v8f c) {
  // V_WMMA_F32_16X16X4_F32 : D = A(16x4) * B(4x16) + C(16x16), all fp32
  return __builtin_amdgcn_wmma_f32_16x16x4_f32(false, a, false, b, (short)0, c,
                                               false, false);
}

// Async global -> LDS copy (CDNA5, ASYNCcnt-tracked). GVS addressing:
// saddr = 64-bit base (SGPR pair), vaddr = per-lane 32-bit byte offset,
// vdst = LDS byte address.
static __device__ __forceinline__ void async_load_b128(unsigned lds_off,
                                                       const float* base,
                                                       unsigned byte_off) {
  asm volatile("global_load_async_to_lds_b128 %0, %1, %2"
               :
               : "v"(lds_off), "v"(byte_off), "s"(base)
               : "memory");
}

#define S_WAIT_ASYNCCNT(n) asm volatile("s_wait_asynccnt " #n ::: "memory")

// LDS byte offset of a generic pointer to a __shared__ object (flat LDS
// aperture keeps the LDS offset in addr[31:0]).
static __device__ __forceinline__ unsigned lds_addr(const void* p) {
  return (unsigned)(unsigned long long)p;
}

static __device__ __forceinline__ float halfmax(float v) {
  v = fmaxf(v, __shfl_xor(v, 1, 32));
  v = fmaxf(v, __shfl_xor(v, 2, 32));
  v = fmaxf(v, __shfl_xor(v, 4, 32));
  v = fmaxf(v, __shfl_xor(v, 8, 32));
  return v;
}
static __device__ __forceinline__ float halfsum(float v) {
  v += __shfl_xor(v, 1, 32);
  v += __shfl_xor(v, 2, 32);
  v += __shfl_xor(v, 4, 32);
  v += __shfl_xor(v, 8, 32);
  return v;
}

// ---------------------------------------------------------------------------
// Kernel A: fused QKV projection.
//   C[4096 x 1536] = x[4096 x 1024] @ [Wq | Wkv]
// Each wave computes one 16x16 tile with v_wmma_f32_16x16x4_f32 (K-loop 256).
// 4 waves per block share the same 16-row x strip; x is staged through LDS in
// 64-wide K chunks, double-buffered with async-to-LDS loads so the copy of
// chunk c+1 overlaps the 16 WMMAs of chunk c. Epilogue folds
// k_scale * DIM_HEAD^-0.5 into Q and scatters Q -> [b][h][n][64],
// K/V -> [b][kh][n][64].
// ---------------------------------------------------------------------------
__global__ __launch_bounds__(128) void qkv_gemm_kernel(
    const float* __restrict__ x, const float* __restrict__ Wq,
    const float* __restrict__ Wkv, const float* __restrict__ k_scale,
    float* __restrict__ Qbuf, float* __restrict__ Kbuf,
    float* __restrict__ Vbuf) {
  __shared__ float Xlds[2][16 * 64];

  const int tid   = threadIdx.x;
  const int wave  = tid >> 5;
  const int lane  = tid & 31;
  const int r     = lane & 15;   // A row / B col / C col
  const int khalf = lane >> 4;   // selects K pair {0,1} vs {2,3}

  const int mt = blockIdx.x;                 // 0..255 : 16-row tile of B*N
  const int ct = blockIdx.y * 4 + wave;      // 0..95  : 16-col tile
  const int m0 = mt * 16;
  const int c0 = ct * 16;

  const bool   isQ = (c0 < QCOLS);
  const float* Wp  = isQ ? Wq : Wkv;
  const int    ldw = isQ ? QCOLS : KVCOLS;
  const int    col = (isQ ? c0 : (c0 - QCOLS)) + r;

  const unsigned xbuf[2] = {lds_addr(&Xlds[0][0]), lds_addr(&Xlds[1][0])};

  // issue async copy of chunk 0: x[m0..m0+15][0..63] (256 float4, 2/thread)
#pragma unroll
  for (int q = 0; q < 2; ++q) {
    const int f = tid + 128 * q;                 // float4 index
    const int row = f >> 4, c4 = f & 15;
    async_load_b128(xbuf[0] + (unsigned)(f * 16), x,
                    (unsigned)(((m0 + row) * DIM + c4 * 4) * 4));
  }

  v8f acc = {};

  for (int c = 0; c < DIM / 64; ++c) {
    const int kc  = c * 64;
    const int cur = c & 1;
    if (c + 1 < DIM / 64) {
      // prefetch next chunk into the other buffer, then wait for current
#pragma unroll
      for (int q = 0; q < 2; ++q) {
        const int f = tid + 128 * q;
        const int row = f >> 4, c4 = f & 15;
        async_load_b128(xbuf[cur ^ 1] + (unsigned)(f * 16), x,
                        (unsigned)(((m0 + row) * DIM + kc + 64 + c4 * 4) * 4));
      }
      S_WAIT_ASYNCCNT(2);   // in-order: current chunk's 2 loads have landed
    } else {
      S_WAIT_ASYNCCNT(0);
    }
    __syncthreads();

    const float* Xc = &Xlds[cur][0];
#pragma unroll
    for (int kk = 0; kk < 64; kk += 4) {
      v2f a, bb;
      a.x = Xc[r * 64 + kk + 2 * khalf];
      a.y = Xc[r * 64 + kk + 2 * khalf + 1];
      const int kg = kc + kk + 2 * khalf;
      bb.x = Wp[(size_t)kg * ldw + col];
      bb.y = Wp[(size_t)(kg + 1) * ldw + col];
      acc = wmma_f32(a, bb, acc);
    }
    __syncthreads();   // everyone done with this buffer before it is refilled
  }

  // Epilogue: C layout -> lane holds rows v+8*khalf, column c0+r.
#pragma unroll
  for (int v = 0; v < 8; ++v) {
    const int row = m0 + v + 8 * khalf;
    const int b   = row >> 11;          // / SEQ
    const int n   = row & (SEQ - 1);
    float val = acc[v];
    const int oc = c0 + r;
    if (oc < QCOLS) {
      const int h = oc >> 6, d = oc & 63;
      val *= k_scale[oc] * 0.125f;      // qk-norm scale + DIM_HEAD^-0.5
      Qbuf[((size_t)(b * HEADS + h) * SEQ + n) * DHEAD + d] = val;
    } else if (oc < QCOLS + KV_HEADS * DHEAD) {
      const int cc = oc - QCOLS;
      const int kh = cc >> 6, d = cc & 63;
      Kbuf[((size_t)(b * KV_HEADS + kh) * SEQ + n) * DHEAD + d] = val;
    } else {
      const int cc = oc - QCOLS - KV_HEADS * DHEAD;
      const int kh = cc >> 6, d = cc & 63;
      Vbuf[((size_t)(b * KV_HEADS + kh) * SEQ + n) * DHEAD + d] = val;
    }
  }
}

// ---------------------------------------------------------------------------
// Kernel B: flash attention over head-major Q/K/V.
// Block = 4 waves; wave w owns 16 query rows; block covers 64 rows of one
// (b, head). K/V 16x64 fp32 tiles are staged in LDS with double-buffered
// async-to-LDS loads (tile j+1 copy overlaps tile j compute). Per KV tile:
// 16 WMMA for S = Q K^T, online softmax (half-wave shfl reductions, wave32
// C-layout keeps each row in one 16-lane half), P routed through padded LDS
// for the C->A layout change, 16 WMMA for O += P V.
// ---------------------------------------------------------------------------
__global__ __launch_bounds__(128) void attn_kernel(
    const float* __restrict__ Qbuf, const float* __restrict__ Kbuf,
    const float* __restrict__ Vbuf, float* __restrict__ out) {
  __shared__ float Klds[2][16 * 64];
  __shared__ float Vlds[2][16 * 64];
  __shared__ float Plds[4][16][17];   // padded to dodge bank conflicts

  const int tid   = threadIdx.x;
  const int wave  = tid >> 5;
  const int lane  = tid & 31;
  const int r     = lane & 15;
  const int khalf = lane >> 4;

  const int bh = blockIdx.y;          // b*HEADS + h
  const int b  = bh / HEADS;
  const int h  = bh % HEADS;
  const int kh = h / GQ;              // kv head
  const int m0 = blockIdx.x * 64 + wave * 16;

  const float* Kbase = Kbuf + (size_t)(b * KV_HEADS + kh) * SEQ * DHEAD;
  const float* Vbase = Vbuf + (size_t)(b * KV_HEADS + kh) * SEQ * DHEAD;

  const unsigned kbuf[2] = {lds_addr(&Klds[0][0]), lds_addr(&Klds[1][0])};
  const unsigned vbuf[2] = {lds_addr(&Vlds[0][0]), lds_addr(&Vlds[1][0])};

  // Preload Q fragment (A layout): lane r holds row m0+r, K's == {0,1} mod 4
  // for khalf=0 and {2,3} mod 4 for khalf=1. Scales already folded in.
  const float* Qrow =
      Qbuf + ((size_t)(b * HEADS + h) * SEQ + (m0 + r)) * DHEAD;
  float qa[32];
#pragma unroll
  for (int i = 0; i < 16; ++i) {
    qa[2 * i]     = Qrow[4 * i + 2 * khalf];
    qa[2 * i + 1] = Qrow[4 * i + 2 * khalf + 1];
  }

  // issue async copy of KV tile 0 (each tile: 256 float4 K + 256 float4 V,
  // 2+2 b128 loads per thread)
  {
    const unsigned lo = (unsigned)(tid * 16);
    async_load_b128(kbuf[0] + lo,        Kbase, lo);
    async_load_b128(kbuf[0] + lo + 2048, Kbase, lo + 2048);
    async_load_b128(vbuf[0] + lo,        Vbase, lo);
    async_load_b128(vbuf[0] + lo + 2048, Vbase, lo + 2048);
  }

  v8f o[4] = {};
  float mrow[8], lrow[8];
#pragma unroll
  for (int v = 0; v < 8; ++v) {
    mrow[v] = -__builtin_inff();
    lrow[v] = 0.0f;
  }

  for (int j = 0; j < SEQ / 16; ++j) {
    const int cur = j & 1;
    if (j + 1 < SEQ / 16) {
      // prefetch tile j+1 into the other buffer, then wait for tile j
      const unsigned lo = (unsigned)(tid * 16);
      const unsigned go = (unsigned)((j + 1) * 16 * DHEAD * 4) + lo;
      async_load_b128(kbuf[cur ^ 1] + lo,        Kbase, go);
      async_load_b128(kbuf[cur ^ 1] + lo + 2048, Kbase, go + 2048);
      async_load_b128(vbuf[cur ^ 1] + lo,        Vbase, go);
      async_load_b128(vbuf[cur ^ 1] + lo + 2048, Vbase, go + 2048);
      S_WAIT_ASYNCCNT(4);   // in-order: tile j's 4 loads have landed
    } else {
      S_WAIT_ASYNCCNT(0);
    }
    __syncthreads();

    const float* Kc = &Klds[cur][0];
    const float* Vc = &Vlds[cur][0];

    // ---- S = Q @ K^T (16x16), K depth 64 -> 16 WMMAs ----
    v8f s = {};
#pragma unroll
    for (int i = 0; i < 16; ++i) {
      v2f a, bb;
      a.x = qa[2 * i];
      a.y = qa[2 * i + 1];
      // B[k][c] = K[c][k] : lane col c == r, lane half selects K pair
      bb.x = Kc[r * 64 + 4 * i + 2 * khalf];
      bb.y = Kc[r * 64 + 4 * i + 2 * khalf + 1];
      s = wmma_f32(a, bb, s);
    }

    // ---- online softmax (rows live across 16-lane halves) ----
#pragma unroll
    for (int v = 0; v < 8; ++v) {
      const float sv   = s[v];
      const float rmax = halfmax(sv);
      const float mnew = fmaxf(mrow[v], rmax);
      const float corr = __expf(mrow[v] - mnew);
      const float p    = __expf(sv - mnew);
      const float rsum = halfsum(p);
      lrow[v] = lrow[v] * corr + rsum;
      mrow[v] = mnew;
      o[0][v] *= corr;
      o[1][v] *= corr;
      o[2][v] *= corr;
      o[3][v] *= corr;
      Plds[wave][v + 8 * khalf][r] = p;   // C-layout -> LDS for transpose
    }
    // Intra-wave DS ops are in-order; reads below see the writes above.

    // ---- O += P @ Vtile : K depth 16, 4 d-tiles -> 16 WMMAs ----
#pragma unroll
    for (int ks4 = 0; ks4 < 16; ks4 += 4) {
      v2f a;
      a.x = Plds[wave][r][ks4 + 2 * khalf];
      a.y = Plds[wave][r][ks4 + 2 * khalf + 1];
#pragma unroll
      for (int t = 0; t < 4; ++t) {
        v2f bb;
        bb.x = Vc[(ks4 + 2 * khalf) * 64 + t * 16 + r];
        bb.y = Vc[(ks4 + 2 * khalf + 1) * 64 + t * 16 + r];
        o[t] = wmma_f32(a, bb, o[t]);
      }
    }
    __syncthreads();   // everyone done with this buffer before it is refilled
  }

  // ---- finalize: out[b][n][h*64 + d] = O / l ----
#pragma unroll
  for (int v = 0; v < 8; ++v) {
    const int   row = m0 + v + 8 * khalf;
    const float inv = 1.0f / lrow[v];
    const size_t base = ((size_t)b * SEQ + row) * QCOLS + h * DHEAD;
#pragma unroll
    for (int t = 0; t < 4; ++t) {
      out[base + t * 16 + r] = o[t][v] * inv;
    }
  }
}

// ---------------------------------------------------------------------------
extern "C" void kernel_launch(void* const* d_in, const int* in_sizes, int n_in,
                              void* d_out, int out_size, void* d_ws,
                              size_t ws_size, hipStream_t stream) {
  const float* x   = (const float*)d_in[0];
  const float* Wq  = (const float*)d_in[1];
  const float* Wkv = (const float*)d_in[2];
  const float* ksc = (const float*)d_in[3];
  float* out = (float*)d_out;

  // workspace: Q [2][16][2048][64], K [2][4][2048][64], V [2][4][2048][64]
  float* Qbuf = (float*)d_ws;
  float* Kbuf = Qbuf + (size_t)BATCH * HEADS * SEQ * DHEAD;
  float* Vbuf = Kbuf + (size_t)BATCH * KV_HEADS * SEQ * DHEAD;

  dim3 gridA(BATCH * SEQ / 16, TOTCOLS / 16 / 4);   // 256 x 24
  qkv_gemm_kernel<<<gridA, 128, 0, stream>>>(x, Wq, Wkv, ksc, Qbuf, Kbuf,
                                             Vbuf);

  dim3 gridB(SEQ / 64, BATCH * HEADS);              // 32 x 32
  attn_kernel<<<gridB, 128, 0, stream>>>(Qbuf, Kbuf, Vbuf, out);
}